// GaussianClockLightGCN_26989574488330
// MI455X (gfx1250) — compile-verified
//
#include <hip/hip_runtime.h>
#include <math.h>

typedef __attribute__((ext_vector_type(2))) float v2f;
typedef __attribute__((ext_vector_type(8))) float v8f;

#define N_USERS   100000
#define M_ITEMS   50000
#define N_NODES   (N_USERS + M_ITEMS)
#define DIM       64
#define KDIM      128          // concatenated K: 64 graph + 64 clock
#define TIME_BINS 24
#define TOPK      3
#define NNZ       4800000
#define BATCH     1024
#define ALPHA     0.5f
#define N_LAYERS  3
#define COL_TILES (M_ITEMS / 16)   // 3125 (exact)
#define ROW_TILES (BATCH / 16)     // 64   (exact)
#define TWO_PI_F  6.28318530717958647692f

// ---------------- init: e0 = concat(user_emb, item_emb); acc = e0 ----------------
__global__ void k_init(const float* __restrict__ ue, const float* __restrict__ ie,
                       float* __restrict__ e0, float* __restrict__ acc) {
    size_t i = (size_t)blockIdx.x * blockDim.x + threadIdx.x;
    size_t n = (size_t)N_NODES * DIM;
    if (i >= n) return;
    size_t ulim = (size_t)N_USERS * DIM;
    float v = (i < ulim) ? ue[i] : ie[i - ulim];
    e0[i] = v;
    acc[i] = v;
}

__global__ void k_zero(float* __restrict__ p, size_t n) {
    size_t i = (size_t)blockIdx.x * blockDim.x + threadIdx.x;
    if (i < n) p[i] = 0.0f;
}

// ---------------- COO SpMM: enext[row] += val * ecur[col], 16 lanes x float4 per nnz ----
__global__ void k_spmm(const float* __restrict__ vals, const int* __restrict__ rows,
                       const int* __restrict__ cols, const float* __restrict__ ecur,
                       float* __restrict__ enext) {
    size_t t = (size_t)blockIdx.x * blockDim.x + threadIdx.x;
    size_t total = (size_t)NNZ * (DIM / 4);      // 16 chunks of 4 floats per nnz
    if (t >= total) return;
    size_t nz = t >> 4;
    int ch = (int)(t & 15);
    int r = rows[nz];
    int c = cols[nz];
    float v = vals[nz];
    const float4 x = *reinterpret_cast<const float4*>(ecur + (size_t)c * DIM + ch * 4);
    float* dst = enext + (size_t)r * DIM + ch * 4;
    atomicAdd(dst + 0, v * x.x);
    atomicAdd(dst + 1, v * x.y);
    atomicAdd(dst + 2, v * x.z);
    atomicAdd(dst + 3, v * x.w);
}

__global__ void k_accum(float* __restrict__ acc, const float* __restrict__ e, size_t n) {
    size_t i = (size_t)blockIdx.x * blockDim.x + threadIdx.x;
    if (i < n) acc[i] += e[i];
}

// ---------------- pack A = [0.25*acc[user] | alpha * v_clock]  [BATCH x 128] ----------
__global__ void k_buildA(const float* __restrict__ acc, const float* __restrict__ cat_emb,
                         const int* __restrict__ top3, const int* __restrict__ users,
                         const float* __restrict__ thetas, float* __restrict__ Amat) {
    int t = blockIdx.x * blockDim.x + threadIdx.x;
    if (t >= BATCH * DIM) return;
    int b = t / DIM;
    int d = t % DIM;
    int u = users[b];
    // graph half (layer mean /4 folded in, symmetrically with B side)
    Amat[(size_t)b * KDIM + d] = 0.25f * acc[(size_t)u * DIM + d];
    // interest-clock half
    float cur = thetas[b] * ((float)TIME_BINS / TWO_PI_F);
    const int* t3 = top3 + (size_t)u * (TIME_BINS * TOPK);
    float num = 0.0f, den = 0.0f;
    #pragma unroll
    for (int h = 0; h < TIME_BINS; ++h) {
        float diff  = fabsf(cur - (float)h);
        float delta = fminf(diff, (float)TIME_BINS - diff);
        float w     = __expf(-0.5f * delta * delta);   // mu=0, sigma=1
        den += w;
        float s = cat_emb[(size_t)t3[h * 3 + 0] * DIM + d]
                + cat_emb[(size_t)t3[h * 3 + 1] * DIM + d]
                + cat_emb[(size_t)t3[h * 3 + 2] * DIM + d];
        num += w * (s * (1.0f / 3.0f));
    }
    Amat[(size_t)b * KDIM + DIM + d] = ALPHA * num / (den + 1e-8f);
}

// ---------------- pack B = [0.25*acc[item] | cat_emb[item_cat]]  [M_ITEMS x 128] ------
__global__ void k_buildB(const float* __restrict__ acc, const float* __restrict__ cat_emb,
                         const int* __restrict__ icat, float* __restrict__ Bmat) {
    size_t t = (size_t)blockIdx.x * blockDim.x + threadIdx.x;
    if (t >= (size_t)M_ITEMS * DIM) return;
    int j = (int)(t / DIM);
    int d = (int)(t % DIM);
    Bmat[(size_t)j * KDIM + d]       = 0.25f * acc[(size_t)(N_USERS + j) * DIM + d];
    Bmat[(size_t)j * KDIM + DIM + d] = cat_emb[(size_t)icat[j] * DIM + d];
}

// ---------------- WMMA GEMM: out[1024 x 50000] = A[1024x128] * B[50000x128]^T --------
// One wave32 computes one 16x16 output tile with 32x V_WMMA_F32_16X16X4_F32.
// A frag (16x4 f32): lanes 0-15 rows, VGPR0/1 = {K, K+1}; lanes 16-31 = {K+2, K+3}.
// B frag (4x16 f32): symmetric by column. Both are contiguous float2 loads from
// the row-major [*,128] packed matrices.
__global__ __launch_bounds__(256) void k_gemm(const float* __restrict__ Amat,
                                              const float* __restrict__ Bmat,
                                              float* __restrict__ out) {
    int wave = threadIdx.x >> 5;
    int lane = threadIdx.x & 31;
    int colTile = blockIdx.x * 8 + wave;
    int rowTile = blockIdx.y;
    if (colTile >= COL_TILES) return;           // wave-uniform: EXEC all-1s inside

    int half = lane >> 4;                        // K-half select
    int idx  = lane & 15;                        // row (A) / col (B) within tile

    const float* Arow = Amat + (size_t)(rowTile * 16 + idx) * KDIM + 2 * half;
    const float* Brow = Bmat + (size_t)(colTile * 16 + idx) * KDIM + 2 * half;

    v8f c = {0.f, 0.f, 0.f, 0.f, 0.f, 0.f, 0.f, 0.f};
    #pragma unroll
    for (int k0 = 0; k0 < KDIM; k0 += 4) {
        v2f a = *reinterpret_cast<const v2f*>(Arow + k0);
        v2f b = *reinterpret_cast<const v2f*>(Brow + k0);
        c = __builtin_amdgcn_wmma_f32_16x16x4_f32(
                /*neg_a=*/false, a, /*neg_b=*/false, b,
                /*c_mod=*/(short)0, c, /*reuse_a=*/false, /*reuse_b=*/false);
    }

    // C/D layout: VGPR v, lane -> row = v + 8*half, col = idx
    int row0 = rowTile * 16 + 8 * half;
    int col  = colTile * 16 + idx;
    #pragma unroll
    for (int v = 0; v < 8; ++v) {
        out[(size_t)(row0 + v) * M_ITEMS + col] = c[v];
    }
}

extern "C" void kernel_launch(void* const* d_in, const int* in_sizes, int n_in,
                              void* d_out, int out_size, void* d_ws, size_t ws_size,
                              hipStream_t stream) {
    const float* user_emb = (const float*)d_in[0];
    const float* item_emb = (const float*)d_in[1];
    const float* cat_emb  = (const float*)d_in[2];
    const float* gvals    = (const float*)d_in[3];
    const int*   grows    = (const int*)d_in[4];
    const int*   gcols    = (const int*)d_in[5];
    const int*   top3     = (const int*)d_in[6];
    const int*   icat     = (const int*)d_in[7];
    const int*   users    = (const int*)d_in[8];
    const float* thetas   = (const float*)d_in[9];
    float* out = (float*)d_out;

    // workspace layout (floats): acc | eA | eB | Amat | Bmat  (~141.6 MB total)
    float* ws   = (float*)d_ws;
    float* acc  = ws;
    float* eA   = acc + (size_t)N_NODES * DIM;
    float* eB   = eA  + (size_t)N_NODES * DIM;
    float* Amat = eB  + (size_t)N_NODES * DIM;
    float* Bmat = Amat + (size_t)BATCH * KDIM;

    const int TB = 256;
    size_t nelem = (size_t)N_NODES * DIM;
    size_t nblk  = (nelem + TB - 1) / TB;

    k_init<<<(unsigned)nblk, TB, 0, stream>>>(user_emb, item_emb, eA, acc);

    float* cur = eA;
    float* nxt = eB;
    for (int l = 0; l < N_LAYERS; ++l) {
        k_zero<<<(unsigned)nblk, TB, 0, stream>>>(nxt, nelem);
        size_t spthreads = (size_t)NNZ * (DIM / 4);
        k_spmm<<<(unsigned)((spthreads + TB - 1) / TB), TB, 0, stream>>>(
            gvals, grows, gcols, cur, nxt);
        k_accum<<<(unsigned)nblk, TB, 0, stream>>>(acc, nxt, nelem);
        float* tmp = cur; cur = nxt; nxt = tmp;
    }

    k_buildA<<<(BATCH * DIM + TB - 1) / TB, TB, 0, stream>>>(
        acc, cat_emb, top3, users, thetas, Amat);
    k_buildB<<<(unsigned)(((size_t)M_ITEMS * DIM + TB - 1) / TB), TB, 0, stream>>>(
        acc, cat_emb, icat, Bmat);

    dim3 grid((COL_TILES + 7) / 8, ROW_TILES);   // 391 x 64 blocks, 8 waves/block
    k_gemm<<<grid, 256, 0, stream>>>(Amat, Bmat, out);
}